// Non_local_46626164966052
// MI455X (gfx1250) — compile-verified
//
#include <hip/hip_runtime.h>
#include <math.h>

// Problem constants (from the reference):
//   B=16, C=512, H=96, W=48, N=H*W=4608, SE_C=32, BN_EPS=1e-5
#define BB 16
#define CC 512
#define NN 4608          // 96*48
#define NQ 1152          // NN/4 float4 per (b,c) row; 1152 = 6*192
#define SEC 32
#define NBLK_P 6         // proj blocks per batch: NQ / 192
#define TPB 192          // 6 wave32 per block for streaming kernels

typedef float v2f __attribute__((ext_vector_type(2)));
typedef float v8f __attribute__((ext_vector_type(8)));

// Deterministic 192 -> 1 LDS tree reduction (fold 192->128, then halve).
__device__ __forceinline__ float block_reduce_192(float v, float* sd) {
    const int t = threadIdx.x;
    sd[t] = v;
    __syncthreads();
    if (t < 64) sd[t] += sd[t + 128];
    __syncthreads();
    for (int off = 64; off > 0; off >>= 1) {
        if (t < off) sd[t] += sd[t + off];
        __syncthreads();
    }
    return sd[0];
}

// ---------------------------------------------------------------------------
// Kernel 1: per-(b,c) spatial mean.  grid = B*C blocks, 192 threads,
// float4 (global_load_b128) streaming: 6 vec4 loads per thread.
// ---------------------------------------------------------------------------
__global__ __launch_bounds__(TPB) void mean_kernel(const float* __restrict__ x,
                                                   float* __restrict__ xp) {
    __shared__ float sd[TPB];
    const int row = blockIdx.x;                 // b*C + c
    const float4* p = (const float4*)(x + (size_t)row * NN);
    const int t = threadIdx.x;
    float sum = 0.f;
#pragma unroll
    for (int k = 0; k < NQ / TPB; ++k) {
        float4 v = p[t + k * TPB];
        sum += (v.x + v.y) + (v.z + v.w);
    }
    float tot = block_reduce_192(sum, sd);
    if (t == 0) xp[row] = tot * (1.f / (float)NN);
}

// ---------------------------------------------------------------------------
// Kernel 2: SE MLP via V_WMMA_F32_16X16X4_F32 on one wave32, plus per-channel
// BN/affine constants.  Layer 1: (16x512)@(512x32); Layer 2: (16x32)@(32x512).
// A-matrix lane layout (16x4 f32):  m = lane%16, k = 2*(lane/16) + vgpr
// B-matrix lane layout (4x16 f32):  n = lane%16, k = 2*(lane/16) + vgpr
// C/D lane layout (16x16 f32):      m = vgpr + 8*(lane/16), n = lane%16
// ---------------------------------------------------------------------------
__global__ __launch_bounds__(32) void se_wmma_kernel(
    const float* __restrict__ xp,       // (16,512)
    const float* __restrict__ se_w1,    // (32,512) row-major
    const float* __restrict__ se_b1,    // (32)
    const float* __restrict__ se_w2,    // (512,32) row-major
    const float* __restrict__ se_b2,    // (512)
    const float* __restrict__ W_w, const float* __restrict__ W_b,
    const float* __restrict__ bn_gamma, const float* __restrict__ bn_beta,
    const float* __restrict__ bn_mean, const float* __restrict__ bn_var,
    float* __restrict__ gate,           // (16,512)
    float* __restrict__ Aarr,           // (512)
    float* __restrict__ Darr)           // (512)
{
    __shared__ float hid[16 * SEC];     // relu output of layer 1
    const int l = threadIdx.x;          // 0..31
    const int m = l & 15;
    const int hh = l >> 4;
    const int kb = hh * 2;

    // ---- layer 1: hid = relu(xp @ se_w1.T + b1), two 16-wide N tiles ----
    for (int nt = 0; nt < 2; ++nt) {
        v8f acc = {};
        const int n = nt * 16 + m;      // column handled by this lane (B side)
        for (int kk = 0; kk < CC / 4; ++kk) {
            const int k0 = kk * 4 + kb;
            v2f a = *(const v2f*)(xp + m * CC + k0);          // A[m][k0..k0+1]
            v2f b = *(const v2f*)(se_w1 + n * CC + k0);       // B[k][n]=w1[n][k]
            acc = __builtin_amdgcn_wmma_f32_16x16x4_f32(
                false, a, false, b, (short)0, acc, false, false);
        }
        const float bias = se_b1[nt * 16 + m];
#pragma unroll
        for (int v = 0; v < 8; ++v) {
            const int row = v + 8 * hh;
            float hv = acc[v] + bias;
            hid[row * SEC + nt * 16 + m] = hv > 0.f ? hv : 0.f;
        }
    }
    __syncthreads();

    // ---- layer 2: gate = sigmoid(hid @ se_w2.T + b2), 32 N tiles ----
    for (int nt = 0; nt < 32; ++nt) {
        v8f acc = {};
        const int ng = nt * 16 + m;
        for (int kk = 0; kk < SEC / 4; ++kk) {
            const int k0 = kk * 4 + kb;
            v2f a = *(const v2f*)(&hid[m * SEC + k0]);
            v2f b = *(const v2f*)(se_w2 + ng * SEC + k0);
            acc = __builtin_amdgcn_wmma_f32_16x16x4_f32(
                false, a, false, b, (short)0, acc, false, false);
        }
        const float b2 = se_b2[ng];
#pragma unroll
        for (int v = 0; v < 8; ++v) {
            const int row = v + 8 * hh;
            float z = acc[v] + b2;
            gate[row * CC + ng] = 1.f / (1.f + __expf(-z));
        }
    }

    // ---- per-channel fused BN/affine constants ----
    for (int c = l; c < CC; c += 32) {
        float inv = bn_gamma[c] * rsqrtf(bn_var[c] + 1e-5f);
        Aarr[c] = W_w[c] * inv;
        Darr[c] = (W_b[c] - bn_mean[c]) * inv + bn_beta[c];
    }
}

// ---------------------------------------------------------------------------
// Kernel 3: per-position theta/phi/g projections (gate folded into weights).
// Each thread owns 4 consecutive positions (one float4 per channel step).
// grid = (NBLK_P, B), 192 threads.  Stores theta (float4), block-reduces
// sum(phi*g) into per-(b,block) partials.
// ---------------------------------------------------------------------------
__global__ __launch_bounds__(TPB) void proj_kernel(
    const float* __restrict__ x, const float* __restrict__ gate,
    const float* __restrict__ g_w, const float* __restrict__ g_b,
    const float* __restrict__ th_w, const float* __restrict__ th_b,
    const float* __restrict__ ph_w, const float* __restrict__ ph_b,
    float* __restrict__ th_out,         // (B, N)
    float* __restrict__ partial)        // (B, NBLK_P)
{
    __shared__ float sd[TPB];
    const int b = blockIdx.y, blk = blockIdx.x, t = threadIdx.x;
    const int q = blk * TPB + t;                 // float4 index within batch row
    const float4* xb4 = (const float4*)(x + (size_t)b * CC * NN) + q;
    const float* gb = gate + b * CC;

    float4 ta = {0.f, 0.f, 0.f, 0.f};
    float4 pa = {0.f, 0.f, 0.f, 0.f};
    float4 ga = {0.f, 0.f, 0.f, 0.f};
    for (int c = 0; c < CC; ++c) {
        if (c + 8 < CC) __builtin_prefetch(xb4 + (size_t)(c + 8) * NQ, 0, 0);
        const float4 xv = xb4[(size_t)c * NQ];
        const float gt = gb[c];                  // uniform across the wave
        const float wt = gt * th_w[c];
        const float wp = gt * ph_w[c];
        const float wg = gt * g_w[c];
        ta.x = fmaf(xv.x, wt, ta.x); ta.y = fmaf(xv.y, wt, ta.y);
        ta.z = fmaf(xv.z, wt, ta.z); ta.w = fmaf(xv.w, wt, ta.w);
        pa.x = fmaf(xv.x, wp, pa.x); pa.y = fmaf(xv.y, wp, pa.y);
        pa.z = fmaf(xv.z, wp, pa.z); pa.w = fmaf(xv.w, wp, pa.w);
        ga.x = fmaf(xv.x, wg, ga.x); ga.y = fmaf(xv.y, wg, ga.y);
        ga.z = fmaf(xv.z, wg, ga.z); ga.w = fmaf(xv.w, wg, ga.w);
    }
    const float tb = th_b[0], pb = ph_b[0], gbias = g_b[0];
    float4 tv = {ta.x + tb, ta.y + tb, ta.z + tb, ta.w + tb};
    ((float4*)(th_out + b * NN))[q] = tv;

    const float pg = (pa.x + pb) * (ga.x + gbias) + (pa.y + pb) * (ga.y + gbias)
                   + (pa.z + pb) * (ga.z + gbias) + (pa.w + pb) * (ga.w + gbias);
    float tot = block_reduce_192(pg, sd);
    if (t == 0) partial[b * NBLK_P + blk] = tot;
}

// ---------------------------------------------------------------------------
// Kernel 4: deterministic final reduction -> s[b] = sum(ph*g)/N
// ---------------------------------------------------------------------------
__global__ __launch_bounds__(32) void sred_kernel(const float* __restrict__ partial,
                                                  float* __restrict__ s) {
    const int b = threadIdx.x;
    if (b < BB) {
        float acc = 0.f;
        for (int j = 0; j < NBLK_P; ++j) acc += partial[b * NBLK_P + j];
        s[b] = acc * (1.f / (float)NN);
    }
}

// ---------------------------------------------------------------------------
// Kernel 5: out = th[b,p]*s[b]*A[c] + D[c] + x[b,c,p]
// grid = (C, B) blocks, 192 threads, float4 streaming (6 vec4 per thread).
// ---------------------------------------------------------------------------
__global__ __launch_bounds__(TPB) void final_kernel(
    const float* __restrict__ x, const float* __restrict__ th,
    const float* __restrict__ s, const float* __restrict__ Aarr,
    const float* __restrict__ Darr, float* __restrict__ out) {
    const int c = blockIdx.x, b = blockIdx.y, t = threadIdx.x;
    const size_t rowoff = ((size_t)(b * CC + c)) * NN;
    const float4* x4 = (const float4*)(x + rowoff);
    const float4* th4 = (const float4*)(th + (size_t)b * NN);
    float4* o4 = (float4*)(out + rowoff);
    const float coef = s[b] * Aarr[c];
    const float d = Darr[c];
#pragma unroll
    for (int k = 0; k < NQ / TPB; ++k) {
        const int q = t + k * TPB;
        const float4 xv = x4[q];
        const float4 tv = th4[q];
        float4 r;
        r.x = fmaf(tv.x, coef, d) + xv.x;
        r.y = fmaf(tv.y, coef, d) + xv.y;
        r.z = fmaf(tv.z, coef, d) + xv.z;
        r.w = fmaf(tv.w, coef, d) + xv.w;
        o4[q] = r;
    }
}

// ---------------------------------------------------------------------------
extern "C" void kernel_launch(void* const* d_in, const int* in_sizes, int n_in,
                              void* d_out, int out_size, void* d_ws, size_t ws_size,
                              hipStream_t stream) {
    (void)in_sizes; (void)n_in; (void)out_size; (void)ws_size;
    const float* x       = (const float*)d_in[0];
    const float* se_w1   = (const float*)d_in[1];
    const float* se_b1   = (const float*)d_in[2];
    const float* se_w2   = (const float*)d_in[3];
    const float* se_b2   = (const float*)d_in[4];
    const float* g_w     = (const float*)d_in[5];
    const float* g_b     = (const float*)d_in[6];
    const float* th_w    = (const float*)d_in[7];
    const float* th_b    = (const float*)d_in[8];
    const float* ph_w    = (const float*)d_in[9];
    const float* ph_b    = (const float*)d_in[10];
    const float* W_w     = (const float*)d_in[11];
    const float* W_b     = (const float*)d_in[12];
    const float* bn_g    = (const float*)d_in[13];
    const float* bn_b    = (const float*)d_in[14];
    const float* bn_m    = (const float*)d_in[15];
    const float* bn_v    = (const float*)d_in[16];
    float* out = (float*)d_out;

    // Workspace layout (floats); total ~360 KB.  All chunks 16B-aligned.
    float* ws      = (float*)d_ws;
    float* xp      = ws;                          // 16*512
    float* gate    = ws + 8192;                   // 16*512
    float* Aarr    = ws + 16384;                  // 512
    float* Darr    = ws + 16896;                  // 512
    float* th      = ws + 17408;                  // 16*4608
    float* partial = ws + 17408 + BB * NN;        // 16*6
    float* sbuf    = partial + BB * NBLK_P;       // 16

    mean_kernel<<<dim3(BB * CC), TPB, 0, stream>>>(x, xp);

    se_wmma_kernel<<<dim3(1), 32, 0, stream>>>(
        xp, se_w1, se_b1, se_w2, se_b2, W_w, W_b, bn_g, bn_b, bn_m, bn_v,
        gate, Aarr, Darr);

    proj_kernel<<<dim3(NBLK_P, BB), TPB, 0, stream>>>(
        x, gate, g_w, g_b, th_w, th_b, ph_w, ph_b, th, partial);

    sred_kernel<<<dim3(1), 32, 0, stream>>>(partial, sbuf);

    final_kernel<<<dim3(CC, BB), TPB, 0, stream>>>(
        x, th, sbuf, Aarr, Darr, out);
}